// GraphAttentionLayer_481036337930
// MI455X (gfx1250) — compile-verified
//
#include <hip/hip_runtime.h>
#include <hip/hip_bf16.h>

// GAT layer for MI455X (gfx1250), wave32 + WMMA.
// Phases:
//   1) h = X @ W^T  via V_WMMA_F32_16X16X4_F32 (full fp32 precision)
//   2) alpha_src/alpha_tgt per (node, head) + stat init
//   3) edge pass A: leaky-relu score -> atomic int-max row_max, deg count
//   4) edge pass B: atomic f32 add of exp(score - row_max)
//   5) denom: 1 / (sum + (N - deg) * exp(-row_max))
//   6) init out: deg>0 ? 0 : h
//   7) edge pass C: out[src] += attn * h[tgt]  (float4 gather, f32 atomics)

#define NN    50000
#define E_CNT 800000
#define K_IN  256
#define OUTC  256
#define HEADS 8
#define DH    32

typedef float v2f __attribute__((ext_vector_type(2)));
typedef float v4f __attribute__((ext_vector_type(4)));
typedef float v8f __attribute__((ext_vector_type(8)));

// ---------------------------------------------------------------------------
// 1) GEMM: Hout[n,o] = sum_k X[n,k] * W[o,k]   (50000x256 @ 256x256^T)
//    Block = 256 threads (8 waves), handles 16 rows x 256 cols.
//    Wave w owns cols [32w, 32w+32) as two 16x16 WMMA tiles.
// ---------------------------------------------------------------------------
__global__ __launch_bounds__(256) void gat_gemm_wmma(
    const float* __restrict__ X, const float* __restrict__ W,
    float* __restrict__ Hout)
{
    __shared__ float ldsX[16][260];   // 260-dword row stride: 16B aligned, bank-rotated

    const int n0  = blockIdx.x * 16;
    const int tid = threadIdx.x;

    // Cooperative coalesced load of the 16x256 X tile (1024 float4's).
    for (int j = 0; j < 4; ++j) {
        int i    = tid + 256 * j;       // float4 index
        int flat = i * 4;
        int row  = flat >> 8;           // / 256
        int col  = flat & 255;
        v4f v = *(const v4f*)(X + (size_t)(n0 + row) * K_IN + col);
        *(v4f*)(&ldsX[row][col]) = v;
    }
    __syncthreads();

    const int lane  = tid & 31;
    const int wave  = tid >> 5;
    const int o0    = wave * 32;
    const int mrow  = lane & 15;          // M (and B's N) index within tile
    const int khalf = (lane >> 4) * 2;    // lanes 16-31 hold K+2,K+3

    v8f c0 = {};
    v8f c1 = {};

    for (int k = 0; k < K_IN; k += 4) {
        // A fragment (16x4 f32): lane holds A[mrow][k+khalf .. +1]
        v2f afrag = *(const v2f*)(&ldsX[mrow][k + khalf]);
        // B fragment (4x16 f32): B[kk][o] = W[o][kk]
        const float* wp0 = W + (size_t)(o0 + mrow) * K_IN + k + khalf;
        v2f b0 = *(const v2f*)wp0;
        v2f b1 = *(const v2f*)(wp0 + 16 * K_IN);
        // 8 args: (neg_a, A, neg_b, B, c_mod, C, reuse_a, reuse_b)
        c0 = __builtin_amdgcn_wmma_f32_16x16x4_f32(
                 false, afrag, false, b0, (short)0, c0, false, false);
        c1 = __builtin_amdgcn_wmma_f32_16x16x4_f32(
                 false, afrag, false, b1, (short)0, c1, false, false);
    }

    // D layout: VGPR r -> row (r + 8*(lane>=16)), col = lane%16
    const int rbase = (lane >> 4) * 8;
    const int col   = lane & 15;
    for (int r = 0; r < 8; ++r) {
        size_t row = (size_t)(n0 + rbase + r);
        Hout[row * OUTC + o0 + col]      = c0[r];
        Hout[row * OUTC + o0 + 16 + col] = c1[r];
    }
}

// ---------------------------------------------------------------------------
// 2) alpha_src/alpha_tgt per (n,h); zero row_max/sum/deg.
// ---------------------------------------------------------------------------
__global__ __launch_bounds__(256) void gat_alpha_init(
    const float* __restrict__ Hf, const float* __restrict__ a,
    float* __restrict__ as, float* __restrict__ at,
    float* __restrict__ rowmax, float* __restrict__ rsum,
    int* __restrict__ deg)
{
    int tid = blockIdx.x * 256 + threadIdx.x;
    if (tid < NN) deg[tid] = 0;
    if (tid >= NN * HEADS) return;
    int n = tid / HEADS;
    int h = tid % HEADS;
    const float* hp = Hf + (size_t)n * OUTC + h * DH;
    float s0 = 0.f, s1 = 0.f;
    #pragma unroll 4
    for (int d = 0; d < DH; ++d) {
        float v = hp[d];
        s0 += v * a[d];
        s1 += v * a[DH + d];
    }
    as[tid] = s0;
    at[tid] = s1;
    rowmax[tid] = 0.f;
    rsum[tid]   = 0.f;
}

__device__ __forceinline__ float leaky(float x) {
    return x >= 0.f ? x : 0.2f * x;
}

// ---------------------------------------------------------------------------
// 3) Edge pass A: row_max (int-bitwise atomic max of positive scores) + deg.
// ---------------------------------------------------------------------------
__global__ __launch_bounds__(256) void gat_edge_max(
    const int* __restrict__ ei,
    const float* __restrict__ as, const float* __restrict__ at,
    float* __restrict__ rowmax, int* __restrict__ deg)
{
    int tid = blockIdx.x * 256 + threadIdx.x;   // exactly E*H threads
    int e = tid >> 3;
    int h = tid & 7;
    int s = ei[e];
    int t = ei[E_CNT + e];
    float sc = leaky(as[s * HEADS + h] + at[t * HEADS + h]);
    if (sc > 0.f)
        atomicMax((int*)&rowmax[s * HEADS + h], __float_as_int(sc));
    if (h == 0)
        atomicAdd(&deg[s], 1);
}

// ---------------------------------------------------------------------------
// 4) Edge pass B: sum of exp(score - row_max[src]).
// ---------------------------------------------------------------------------
__global__ __launch_bounds__(256) void gat_edge_sum(
    const int* __restrict__ ei,
    const float* __restrict__ as, const float* __restrict__ at,
    const float* __restrict__ rowmax, float* __restrict__ rsum)
{
    int tid = blockIdx.x * 256 + threadIdx.x;
    int e = tid >> 3;
    int h = tid & 7;
    int s = ei[e];
    int t = ei[E_CNT + e];
    int nh = s * HEADS + h;
    float sc = leaky(as[nh] + at[t * HEADS + h]);
    float ex = __expf(sc - rowmax[nh]);
    atomicAdd(&rsum[nh], ex);
}

// ---------------------------------------------------------------------------
// 5) rsum <- 1 / (rsum + (N - deg) * exp(-row_max))
// ---------------------------------------------------------------------------
__global__ __launch_bounds__(256) void gat_denom(
    const float* __restrict__ rowmax, const int* __restrict__ deg,
    float* __restrict__ rsum)
{
    int tid = blockIdx.x * 256 + threadIdx.x;
    if (tid >= NN * HEADS) return;
    int n = tid / HEADS;
    float denom = rsum[tid] + (float)(NN - deg[n]) * __expf(-rowmax[tid]);
    rsum[tid] = 1.0f / denom;
}

// ---------------------------------------------------------------------------
// 6) out = deg>0 ? 0 : h   (full rewrite every call)
// ---------------------------------------------------------------------------
__global__ __launch_bounds__(256) void gat_init_out(
    const float* __restrict__ Hf, const int* __restrict__ deg,
    float* __restrict__ out)
{
    int tid = blockIdx.x * 256 + threadIdx.x;   // exactly N*256 threads
    int n = tid >> 8;
    out[tid] = (deg[n] > 0) ? 0.f : Hf[tid];
}

// ---------------------------------------------------------------------------
// 7) Edge pass C: out[src] += attn * h[tgt].  64 threads/edge: 8 per head,
//    each handles a float4 channel group.
// ---------------------------------------------------------------------------
__global__ __launch_bounds__(256) void gat_edge_agg(
    const int* __restrict__ ei,
    const float* __restrict__ as, const float* __restrict__ at,
    const float* __restrict__ rowmax, const float* __restrict__ rinv,
    const float* __restrict__ Hf, float* __restrict__ out)
{
    int tid = blockIdx.x * 256 + threadIdx.x;   // exactly E*64 threads
    int e   = tid >> 6;
    int t64 = tid & 63;
    int h   = t64 >> 3;
    int j   = t64 & 7;
    int s = ei[e];
    int t = ei[E_CNT + e];
    int nh = s * HEADS + h;
    float sc   = leaky(as[nh] + at[t * HEADS + h]);
    float attn = __expf(sc - rowmax[nh]) * rinv[nh];
    const float* hp = Hf  + (size_t)t * OUTC + h * DH + j * 4;
    float*       op = out + (size_t)s * OUTC + h * DH + j * 4;
    v4f v = *(const v4f*)hp;
    atomicAdd(op + 0, attn * v.x);
    atomicAdd(op + 1, attn * v.y);
    atomicAdd(op + 2, attn * v.z);
    atomicAdd(op + 3, attn * v.w);
}

// ---------------------------------------------------------------------------
extern "C" void kernel_launch(void* const* d_in, const int* in_sizes, int n_in,
                              void* d_out, int out_size, void* d_ws, size_t ws_size,
                              hipStream_t stream)
{
    const float* X  = (const float*)d_in[0];   // N x 256
    const int*   ei = (const int*)  d_in[1];   // 2 x E (src row, then tgt row)
    const float* W  = (const float*)d_in[2];   // 256 x 256
    const float* a  = (const float*)d_in[3];   // 64 (a_src[0:32], a_tgt[32:64])
    float* out = (float*)d_out;                // N x 256

    // Workspace layout (floats): h | alpha_src | alpha_tgt | row_max | rsum | deg(int)
    float* ws     = (float*)d_ws;
    float* h_ws   = ws;                                   // NN*256
    float* as_ws  = h_ws  + (size_t)NN * OUTC;            // NN*8
    float* at_ws  = as_ws + (size_t)NN * HEADS;           // NN*8
    float* rm_ws  = at_ws + (size_t)NN * HEADS;           // NN*8
    float* rs_ws  = rm_ws + (size_t)NN * HEADS;           // NN*8
    int*   deg_ws = (int*)(rs_ws + (size_t)NN * HEADS);   // NN

    const int nh_blocks = (NN * HEADS + 255) / 256;       // 1563

    // 1) GEMM (WMMA f32): 3125 blocks x 16 rows
    gat_gemm_wmma<<<NN / 16, 256, 0, stream>>>(X, W, h_ws);
    // 2) alpha + stat init
    gat_alpha_init<<<nh_blocks, 256, 0, stream>>>(h_ws, a, as_ws, at_ws, rm_ws, rs_ws, deg_ws);
    // 3) row_max + deg
    gat_edge_max<<<(E_CNT * HEADS) / 256, 256, 0, stream>>>(ei, as_ws, at_ws, rm_ws, deg_ws);
    // 4) sum of exp
    gat_edge_sum<<<(E_CNT * HEADS) / 256, 256, 0, stream>>>(ei, as_ws, at_ws, rm_ws, rs_ws);
    // 5) 1/denom
    gat_denom<<<nh_blocks, 256, 0, stream>>>(rm_ws, deg_ws, rs_ws);
    // 6) init output
    gat_init_out<<<(NN * OUTC) / 256, 256, 0, stream>>>(h_ws, deg_ws, out);
    // 7) aggregate
    gat_edge_agg<<<(E_CNT * 64) / 256, 256, 0, stream>>>(ei, as_ws, at_ws, rm_ws, rs_ws, h_ws, out);
}